// GCN_28638841930491
// MI455X (gfx1250) — compile-verified
//
#include <hip/hip_runtime.h>

typedef __attribute__((ext_vector_type(2))) float v2f;
typedef __attribute__((ext_vector_type(8))) float v8f;

#define FDIM 128

// ---------------- degree / normalization ----------------

__global__ __launch_bounds__(256) void k_deg_init(unsigned* deg, int n) {
    int i = blockIdx.x * 256 + threadIdx.x;
    if (i < n) deg[i] = 1u;  // self loop
}

__global__ __launch_bounds__(256) void k_deg_edges(const int* __restrict__ dst,
                                                   unsigned* deg, int nE) {
    int e = blockIdx.x * 256 + threadIdx.x;
    if (e < nE) atomicAdd(&deg[dst[e]], 1u);
}

__global__ __launch_bounds__(256) void k_dinv(float* buf, int n) {
    int i = blockIdx.x * 256 + threadIdx.x;
    if (i < n) {
        unsigned d = ((const unsigned*)buf)[i];
        buf[i] = rsqrtf((float)d);  // deg >= 1 always (self loop)
    }
}

// ---------------- fp32 WMMA GEMM: H = X @ W  (K = N = 128) ----------------
// One wave computes a 16x16 tile; 8 waves/block cover all 128 output columns
// for one 16-row stripe. W (64KB) is staged in LDS and reused across k-steps.

__global__ __launch_bounds__(256) void k_gemm_wmma(const float* __restrict__ X,
                                                   const float* __restrict__ W,
                                                   float* __restrict__ H, int n) {
    __shared__ float ldsW[FDIM * FDIM];
    {
        const float4* wv = (const float4*)W;
        float4* lv = (float4*)ldsW;
        for (int i = threadIdx.x; i < FDIM * FDIM / 4; i += 256) lv[i] = wv[i];
    }
    __syncthreads();

    const int lane  = threadIdx.x & 31;
    const int wid   = threadIdx.x >> 5;   // wave id 0..7 -> 16-col tile
    const int m0    = blockIdx.x * 16;
    const int n0    = wid * 16;
    const int half  = lane >> 4;          // 0: K pair {0,1}, 1: K pair {2,3}
    const int idx16 = lane & 15;
    const int kb    = half * 2;

    long arow = (long)m0 + idx16;
    if (arow > (long)n - 1) arow = (long)n - 1;   // clamp (n is mult of 16 here)
    const float* aptr = X + arow * FDIM + kb;
    const float* bptr = ldsW + kb * FDIM + n0 + idx16;

    v8f c = {};
#pragma unroll
    for (int k = 0; k < FDIM; k += 4) {
        v2f a = *(const v2f*)(aptr + k);          // A: M=idx16, K=k+kb, k+kb+1
        v2f b;
        b.x = bptr[k * FDIM];                     // B: K=k+kb,   N=n0+idx16
        b.y = bptr[k * FDIM + FDIM];              // B: K=k+kb+1, N=n0+idx16
        c = __builtin_amdgcn_wmma_f32_16x16x4_f32(
                false, a, false, b, (short)0, c, false, false);
    }

    // D: VGPR v -> row m0 + v + 8*half, col n0 + idx16
    const int mbase = m0 + half * 8;
    float* o = H + (long)mbase * FDIM + n0 + idx16;
#pragma unroll
    for (int v = 0; v < 8; ++v)
        if (mbase + v < n) o[(long)v * FDIM] = c[v];
}

// ---------------- self-loop contribution + bias (initializes accumulator) ---

__global__ __launch_bounds__(256) void k_self_bias(const float* __restrict__ H,
                                                   const float* __restrict__ dinv,
                                                   const float* __restrict__ bias,
                                                   float* __restrict__ out, int n) {
    int t = blockIdx.x * 256 + threadIdx.x;
    int i = t >> 5, lane = t & 31;               // 32 lanes x float4 = 128 feats
    if (i >= n) return;
    float di = dinv[i];
    float w  = di * di;
    float4 h = *(const float4*)(H + (long)i * FDIM + lane * 4);
    float4 b = *(const float4*)(bias + lane * 4);
    float4 o = make_float4(h.x * w + b.x, h.y * w + b.y,
                           h.z * w + b.z, h.w * w + b.w);
    *(float4*)(out + (long)i * FDIM + lane * 4) = o;
}

// ---------------- edge scatter: out[dst] += h[src] * dinv[s]*dinv[d] --------

__global__ __launch_bounds__(256) void k_scatter(const int* __restrict__ src,
                                                 const int* __restrict__ dst,
                                                 const float* __restrict__ dinv,
                                                 const float* __restrict__ H,
                                                 float* out, int nE) {
    int t = blockIdx.x * 256 + threadIdx.x;
    int e = t >> 5, lane = t & 31;               // 32 lanes x float4 per edge
    if (e >= nE) return;
    int s = src[e], d = dst[e];
    float w = dinv[s] * dinv[d];
    float4 h = *(const float4*)(H + (long)s * FDIM + lane * 4);
    float* o = out + (long)d * FDIM + lane * 4;
    unsafeAtomicAdd(o + 0, h.x * w);   // global_atomic_add_f32, L2-resident
    unsafeAtomicAdd(o + 1, h.y * w);
    unsafeAtomicAdd(o + 2, h.z * w);
    unsafeAtomicAdd(o + 3, h.w * w);
}

// ---------------- ReLU ------------------------------------------------------

__global__ __launch_bounds__(256) void k_relu(float* buf, long nq) {
    long t = (long)blockIdx.x * 256 + threadIdx.x;
    if (t >= nq) return;
    float4* p = (float4*)buf + t;
    float4 v = *p;
    v.x = fmaxf(v.x, 0.f); v.y = fmaxf(v.y, 0.f);
    v.z = fmaxf(v.z, 0.f); v.w = fmaxf(v.w, 0.f);
    *p = v;
}

// ---------------- driver ----------------------------------------------------

extern "C" void kernel_launch(void* const* d_in, const int* in_sizes, int n_in,
                              void* d_out, int out_size, void* d_ws, size_t ws_size,
                              hipStream_t stream) {
    const float* x  = (const float*)d_in[0];
    const int*   ei = (const int*)d_in[1];
    const float* W1 = (const float*)d_in[2];
    const float* b1 = (const float*)d_in[3];
    const float* W2 = (const float*)d_in[4];
    const float* b2 = (const float*)d_in[5];

    const int n  = in_sizes[0] / FDIM;   // 100000
    const int nE = in_sizes[1] / 2;      // 1600000
    const int* srcp = ei;
    const int* dstp = ei + nE;
    float* out = (float*)d_out;

    float* wsf  = (float*)d_ws;
    float* dinv = wsf;                         // n floats (reserve 131072)
    unsigned* deg = (unsigned*)wsf;            // aliases dinv (in-place rsqrt)
    float* hbuf = wsf + 131072;                // n*128 floats
    float* agg  = hbuf + (long)n * FDIM;       // n*128 floats

    dim3 B(256);
    int nodeB1  = (n + 255) / 256;
    int edgeB1  = (nE + 255) / 256;
    int gemmB   = (n + 15) / 16;
    int nodeB32 = (int)(((long)n * 32 + 255) / 256);
    int edgeB32 = (int)(((long)nE * 32 + 255) / 256);
    long nq     = (long)n * FDIM / 4;
    int reluB   = (int)((nq + 255) / 256);

    // normalization: deg (incl. self loop) -> dinv = rsqrt(deg)
    k_deg_init <<<nodeB1, B, 0, stream>>>(deg, n);
    k_deg_edges<<<edgeB1, B, 0, stream>>>(dstp, deg, nE);
    k_dinv     <<<nodeB1, B, 0, stream>>>(dinv, n);

    // layer 1: h = x@W1 ; agg = dinv^2*h + b1 ; agg[dst] += h[src]*norm ; relu
    k_gemm_wmma<<<gemmB,   B, 0, stream>>>(x, W1, hbuf, n);
    k_self_bias<<<nodeB32, B, 0, stream>>>(hbuf, dinv, b1, agg, n);
    k_scatter  <<<edgeB32, B, 0, stream>>>(srcp, dstp, dinv, hbuf, agg, nE);
    k_relu     <<<reluB,   B, 0, stream>>>(agg, nq);

    // layer 2: h = agg@W2 ; out = dinv^2*h + b2 ; out[dst] += h[src]*norm
    k_gemm_wmma<<<gemmB,   B, 0, stream>>>(agg, W2, hbuf, n);
    k_self_bias<<<nodeB32, B, 0, stream>>>(hbuf, dinv, b2, out, n);
    k_scatter  <<<edgeB32, B, 0, stream>>>(srcp, dstp, dinv, hbuf, out, nE);
}